// SHModule_19164144075137
// MI455X (gfx1250) — compile-verified
//
#include <hip/hip_runtime.h>

#define L_MAX 16
#define NCOL ((L_MAX + 1) * (L_MAX + 1))   // 289 output columns
#define BLK 64                             // 2 waves (wave32) per block

// ---------------- compile-time prefactor table (folds to literals) -------
constexpr double kPI = 3.14159265358979323846264338327950288;

constexpr double csqrt_(double x) {
  double g = (x > 1.0) ? x : 1.0;
  for (int i = 0; i < 200; ++i) {
    double n = 0.5 * (g + x / g);
    if (n == g) break;
    g = n;
  }
  return g;
}

constexpr double cfact_(int n) {
  double r = 1.0;
  for (int i = 2; i <= n; ++i) r *= (double)i;
  return r;
}

struct FTab { float v[L_MAX + 1][L_MAX + 1]; };

constexpr FTab buildF_() {
  FTab t{};
  for (int l = 0; l <= L_MAX; ++l)
    for (int m = 0; m <= l; ++m) {
      const double sgn = (m & 1) ? -1.0 : 1.0;
      t.v[l][m] = (float)(sgn * csqrt_(((2.0 * l + 1.0) / (2.0 * kPI)) *
                                       (cfact_(l - m) / cfact_(l + m))));
    }
  return t;
}

constexpr FTab FT = buildF_();
constexpr float INV_SQRT2 = 0.70710678118654752440084436210485f;

typedef unsigned int u32x4 __attribute__((ext_vector_type(4)));
typedef int i32x4 __attribute__((ext_vector_type(4)));
typedef int i32x8 __attribute__((ext_vector_type(8)));

__global__ __launch_bounds__(BLK) void sh_kernel(const float* __restrict__ xyz,
                                                 float* __restrict__ out,
                                                 int N) {
  // One full output tile staged in LDS: 64 rows x 289 f32 = 73,984 B.
  __shared__ float tile[BLK * NCOL];

  const int tid = threadIdx.x;
  const int n0  = blockIdx.x * BLK;
  const int n   = n0 + tid;
  const int row = tid * NCOL;   // row stride 289 floats -> conflict-free banks

  if (n < N) {
    const float x = xyz[3 * n + 0];
    const float y = xyz[3 * n + 1];
    const float z = xyz[3 * n + 2];
    const float r2 = x * x + y * y + z * z;

    // c/s recurrence (cos/sin of m*phi scaled by rho^m)
    float c[L_MAX + 1], s[L_MAX + 1];
    c[0] = 1.0f;
    s[0] = 0.0f;
#pragma unroll
    for (int m = 1; m <= L_MAX; ++m) {
      c[m] = x * c[m - 1] - y * s[m - 1];
      s[m] = x * s[m - 1] + y * c[m - 1];
    }

    float Qp[L_MAX + 1], Qpp[L_MAX + 1], Qc[L_MAX + 1];

    // l = 0
    Qc[0] = 1.0f;
    tile[row + 0] = FT.v[0][0] * INV_SQRT2;
    Qp[0] = 1.0f;

#pragma unroll
    for (int l = 1; l <= L_MAX; ++l) {
      const float a = (float)(2 * l - 1);
      Qc[l]     = -a * Qp[l - 1];
      Qc[l - 1] =  a * z * Qp[l - 1];
#pragma unroll
      for (int m = l - 2; m >= 0; --m) {
        Qc[m] = (a * z * Qp[m] - (float)(l + m - 1) * r2 * Qpp[m]) *
                (float)(1.0 / (double)(l - m));
      }

      const int base = row + l * l;
#pragma unroll
      for (int mm = 1; mm <= l; ++mm)          // m = -mm  (uses s)
        tile[base + (l - mm)] = FT.v[l][mm] * Qc[mm] * s[mm];
      tile[base + l] = FT.v[l][0] * Qc[0] * INV_SQRT2;   // m = 0
#pragma unroll
      for (int mm = 1; mm <= l; ++mm)          // m = +mm  (uses c)
        tile[base + l + mm] = FT.v[l][mm] * Qc[mm] * c[mm];

      // rotate recurrence state
#pragma unroll
      for (int m = 0; m <= l - 1; ++m) Qpp[m] = Qp[m];
#pragma unroll
      for (int m = 0; m <= l; ++m) Qp[m] = Qc[m];
    }
  }

  __syncthreads();   // all LDS writes visible before the DMA reads them

  const int rows  = (N - n0 < BLK) ? (N - n0) : BLK;  // partial last block
  const int nelem = rows * NCOL;                      // elements to flush

#if defined(__gfx1250__) && __has_builtin(__builtin_amdgcn_tensor_store_from_lds)
  // One TDM tensor-store per block: flat 1-D tile of `nelem` dwords,
  // LDS -> contiguous global rows [n0, n0+rows). Issued by wave 0 only
  // (scalar branch: TDM ignores EXEC, so predication must be wave-level).
  if (__builtin_amdgcn_readfirstlane(tid) == 0) {
    const unsigned lds_base =
        (unsigned)(unsigned long long)(uintptr_t)&tile[0];  // flat low 32 = LDS offset
    const unsigned long long ga =
        (unsigned long long)(uintptr_t)(out + (size_t)n0 * NCOL);
    const unsigned ne = (unsigned)nelem;                    // <= 18,496

    // D# group 0: count=1 | lds_addr | global_addr[56:0] | type=2
    u32x4 g0 = { 1u,
                 lds_base,
                 (unsigned)(ga & 0xFFFFFFFFull),
                 (unsigned)((ga >> 32) & 0x01FFFFFFull) | (2u << 30) };
    // D# group 1: mask=0, data_size=4B, tensor_dim0=ne, tensor_dim1=1,
    //             tile_dim0=ne, tile_dim1=1, tile_dim2=0, dim0_stride=ne
    i32x8 g1 = { (int)(2u << 16),                    // data_size = 2 (4 bytes)
                 (int)((ne & 0xFFFFu) << 16),        // tensor_dim0[15:0]
                 (int)((ne >> 16) | (1u << 16)),     // tensor_dim0[31:16] | tensor_dim1=1
                 (int)((ne & 0xFFFFu) << 16),        // tile_dim0
                 1,                                  // tile_dim1=1 | tile_dim2=0
                 (int)ne,                            // tensor_dim0_stride[31:0]
                 0, 0 };
    i32x4 gz = { 0, 0, 0, 0 };
#if __clang_major__ >= 23
    i32x8 gz8 = { 0, 0, 0, 0, 0, 0, 0, 0 };
    __builtin_amdgcn_tensor_store_from_lds(g0, g1, gz, gz, gz8, 0);
#else
    __builtin_amdgcn_tensor_store_from_lds(g0, g1, gz, gz, 0);
#endif
    __builtin_amdgcn_s_wait_tensorcnt(0);   // keep LDS alive until DMA done
  }
#else
  // Fallback (also host pass): coalesced float4 flush of the flat tile.
  const int nelem4 = nelem >> 2;
  float4* __restrict__ dst4 =
      reinterpret_cast<float4*>(out + (size_t)n0 * NCOL);
  const float4* src4 = reinterpret_cast<const float4*>(tile);
  for (int i = tid; i < nelem4; i += BLK) dst4[i] = src4[i];
  for (int i = (nelem4 << 2) + tid; i < nelem; i += BLK)
    out[(size_t)n0 * NCOL + i] = tile[i];
#endif
}

extern "C" void kernel_launch(void* const* d_in, const int* in_sizes, int n_in,
                              void* d_out, int out_size, void* d_ws, size_t ws_size,
                              hipStream_t stream) {
  (void)n_in; (void)out_size; (void)d_ws; (void)ws_size;
  const float* xyz = (const float*)d_in[0];
  float* out = (float*)d_out;
  const int N = in_sizes[0] / 3;            // (N, 3) flattened
  const int grid = (N + BLK - 1) / BLK;
  sh_kernel<<<grid, BLK, 0, stream>>>(xyz, out, N);
}